// GCNConv_diag_17712445129317
// MI455X (gfx1250) — compile-verified
//
#include <hip/hip_runtime.h>
#include <stdint.h>

#define DF     128   // feature dim (compile-time per reference)
#define CHUNK  256   // edges staged per block
#define BLOCK  256   // threads per block = 8 waves (wave32)

typedef float        float4v __attribute__((ext_vector_type(4)));
typedef unsigned int uint4v  __attribute__((ext_vector_type(4)));

// ---------------- CDNA5 async global->LDS copy (ASYNCcnt) ----------------
__device__ __forceinline__ void async_ld_b32(const void* gptr, void* lptr) {
#if defined(__HIP_DEVICE_COMPILE__)
#if __has_builtin(__builtin_amdgcn_global_load_async_to_lds_b32)
  __builtin_amdgcn_global_load_async_to_lds_b32(
      (__attribute__((address_space(1))) int*)(uintptr_t)gptr,
      (__attribute__((address_space(3))) int*)(uint32_t)(uintptr_t)lptr,
      /*imm offset*/ 0, /*cpol*/ 0);
#else
  unsigned int loff = (unsigned int)(uintptr_t)lptr;  // low 32 bits = LDS offset
  asm volatile("global_load_async_to_lds_b32 %0, %1, off"
               :: "v"(loff), "v"(gptr) : "memory");
#endif
#else
  (void)gptr; (void)lptr;
#endif
}

__device__ __forceinline__ void wait_async0() {
#if defined(__HIP_DEVICE_COMPILE__)
#if __has_builtin(__builtin_amdgcn_s_wait_asynccnt)
  __builtin_amdgcn_s_wait_asynccnt(0);
#else
  asm volatile("s_wait_asynccnt 0" ::: "memory");
#endif
#endif
}

// Native no-return f32 atomic add (resolved at the L2 atomic units;
// out is 5.12 MB and stays L2-resident on the 192 MB L2)
__device__ __forceinline__ void atomic_add_f32(float* p, float v) {
  __hip_atomic_fetch_add(p, v, __ATOMIC_RELAXED, __HIP_MEMORY_SCOPE_AGENT);
}

// ---------------- zero the accumulator (harness poisons d_out) ----------------
__global__ void zero_f32x4_kernel(float4v* __restrict__ out, int n4) {
  int i = blockIdx.x * blockDim.x + threadIdx.x;
  if (i < n4) out[i] = (float4v){0.f, 0.f, 0.f, 0.f};
}

// ---------------- fused diag-scale + COO scatter-add SpMM ----------------
// One wave32 per edge: lane l owns features [4l, 4l+4). Each wave async-stages
// the 32 edges it will consume into 16B-aligned LDS records (row,col,val,pad),
// drains with s_wait_asynccnt 0 (no barrier needed: strictly own data), then
// reads each record back with a single ds_load_b128 broadcast.
__global__ __launch_bounds__(BLOCK)
void gcn_diag_spmm_kernel(const float* __restrict__ x,
                          const float* __restrict__ W,
                          const int*   __restrict__ rows,
                          const int*   __restrict__ cols,
                          const float* __restrict__ vals,
                          float*       __restrict__ out,
                          int E) {
  __shared__ __align__(16) unsigned int s_edge[CHUNK * 4];  // {row,col,val,pad}

  const int tid  = threadIdx.x;
  const int lane = tid & 31;
  const int wave = tid >> 5;                 // 0..7
  const int base = blockIdx.x * CHUNK;

  // Stage this block's edge chunk (12 B/edge) with async LDS copies.
  const int e = base + tid;
  if (e < E) {
    async_ld_b32(rows + e, &s_edge[4 * tid + 0]);
    async_ld_b32(cols + e, &s_edge[4 * tid + 1]);
    async_ld_b32(vals + e, &s_edge[4 * tid + 2]);
  }
  wait_async0();   // each wave waits for its own async ops; it reads only those

  // Per-lane slice of the learned diagonal (reused across all 32 edges).
  const float4v w4 = *(const float4v*)(W + lane * 4);

  const int nEdges = (E - base < CHUNK) ? (E - base) : CHUNK;
  const int lo = wave * 32;
  const int hi = (lo + 32 < nEdges) ? (lo + 32) : nEdges;

#pragma unroll 4
  for (int i = lo; i < hi; ++i) {
    const uint4v rec = *(const uint4v*)(&s_edge[4 * i]);  // one ds_load_b128
    const int   r = (int)rec.x;
    const int   c = (int)rec.y;
    const float v = __uint_as_float(rec.z);

    // 512 B gather per edge: 32 lanes x global_load_b128, 4 coalesced lines.
    const float4v xv = *(const float4v*)(x + (size_t)c * DF + lane * 4);
    const float4v contrib = xv * (w4 * v);

    float* o = out + (size_t)r * DF + lane * 4;
    atomic_add_f32(o + 0, contrib.x);  // no-return global_atomic_add_f32
    atomic_add_f32(o + 1, contrib.y);  // (STOREcnt; fire-and-forget)
    atomic_add_f32(o + 2, contrib.z);
    atomic_add_f32(o + 3, contrib.w);
  }
}

extern "C" void kernel_launch(void* const* d_in, const int* in_sizes, int n_in,
                              void* d_out, int out_size, void* d_ws, size_t ws_size,
                              hipStream_t stream) {
  const float* x    = (const float*)d_in[0];   // [N, 128]
  const float* W    = (const float*)d_in[1];   // [128]
  const int*   rows = (const int*)  d_in[2];   // [E]
  const int*   cols = (const int*)  d_in[3];   // [E]
  const float* vals = (const float*)d_in[4];   // [E]
  float*       out  = (float*)d_out;           // [N, 128]

  const int E  = in_sizes[2];
  const int n4 = out_size / 4;

  zero_f32x4_kernel<<<(n4 + 255) / 256, 256, 0, stream>>>((float4v*)out, n4);

  const int blocks = (E + CHUNK - 1) / CHUNK;  // 2500 blocks, 8 waves each
  gcn_diag_spmm_kernel<<<blocks, BLOCK, 0, stream>>>(x, W, rows, cols, vals,
                                                     out, E);
}